// YOLODetectionModel_8246337208742
// MI455X (gfx1250) — compile-verified
//
#include <hip/hip_runtime.h>
#include <hip/hip_bf16.h>

// ---------------- problem constants ----------------
#define NB    32      // batch
#define NCH   84      // 4 box + 80 classes
#define NA    8400    // anchors
#define NCLS  80
#define NSORT 16384   // pow2 >= NA for bitonic sort
#define TOPK  1000
#define KPAD  1024    // padded NMS set (multiple of 16 for WMMA tiles)
#define NT    (KPAD/16)              // 64 tiles per dim
#define NTILES (NT*(NT+1)/2)         // 2080 upper-tri tiles
#define MAXDET 300
#define CONF_T 0.25f
#define IOU_T  0.7f
#define MAXWH  7680.0f
#define IMGSZ  640.0f

// per-image workspace layout (floats): score[1024] | label[1024](int) | box[1024*4] | offbox[1024*4]
#define WS_PER_IMG 10240

typedef __attribute__((ext_vector_type(2))) float v2f;
typedef __attribute__((ext_vector_type(8))) float v8f;

__device__ __constant__ int d_coco[NCLS] = {
  1,2,3,4,5,6,7,8,9,10,11,13,14,15,16,17,18,19,20,21,22,23,24,25,27,28,31,32,33,34,
  35,36,37,38,39,40,41,42,43,44,46,47,48,49,50,51,52,53,54,55,56,57,58,59,60,61,62,
  63,64,65,67,70,72,73,74,75,76,77,78,79,80,81,82,84,85,86,87,88,89,90};

// ------------------------------------------------------------------
// Kernel 1: per image — fused class-max scoring (b128 vectorized over
// 4 anchors: this is the HBM-roofline pass, ~86MB total), in-LDS
// 16K-entry bitonic sort (128KB of CDNA5's 320KB LDS), then gather
// top-1024 candidates to workspace.
// ------------------------------------------------------------------
__global__ __launch_bounds__(1024) void k_score_sort(const float* __restrict__ preds,
                                                     float* __restrict__ ws) {
  __shared__ float    skey[NSORT];   // 64 KB  (ascending key = -masked_score)
  __shared__ unsigned sval[NSORT];   // 64 KB  (anchor | label<<14)

  const int b   = blockIdx.x;
  const int tid = threadIdx.x;
  const float* __restrict__ P = preds + (size_t)b * NCH * NA;

  // 4 anchors per thread iteration: class rows are 8400*4B (16B multiple)
  // apart, so each read is a full global_load_b128.
  for (int idx = tid; idx < NA / 4; idx += 1024) {
    const int a0 = idx * 4;
    float b0 = -1e30f, b1 = -1e30f, b2 = -1e30f, b3 = -1e30f;
    int   l0 = 0, l1 = 0, l2 = 0, l3 = 0;
    #pragma unroll 4
    for (int c = 0; c < NCLS; ++c) {
      const float4 v = *reinterpret_cast<const float4*>(P + (size_t)(4 + c) * NA + a0);
      if (v.x > b0) { b0 = v.x; l0 = c; }   // first-max == jnp.argmax
      if (v.y > b1) { b1 = v.y; l1 = c; }
      if (v.z > b2) { b2 = v.z; l2 = c; }
      if (v.w > b3) { b3 = v.w; l3 = c; }
    }
    const float bs[4] = {b0, b1, b2, b3};
    const int   ls[4] = {l0, l1, l2, l3};
    #pragma unroll
    for (int e = 0; e < 4; ++e) {
      float m = (bs[e] > CONF_T) ? bs[e] : -1.0f;
      skey[a0 + e] = -m;
      sval[a0 + e] = (unsigned)(a0 + e) | ((unsigned)ls[e] << 14);
    }
  }
  for (int a = NA + tid; a < NSORT; a += 1024) { skey[a] = 1e30f; sval[a] = 0u; }
  __syncthreads();

  // bitonic sort ascending on key (=> descending score)
  for (int k = 2; k <= NSORT; k <<= 1) {
    for (int j = k >> 1; j > 0; j >>= 1) {
      for (int i = tid; i < NSORT; i += 1024) {
        int l = i ^ j;
        if (l > i) {
          bool up = ((i & k) == 0);
          float ki = skey[i], kl = skey[l];
          if ((ki > kl) == up) {
            unsigned vi = sval[i];
            skey[i] = kl; skey[l] = ki;
            sval[i] = sval[l]; sval[l] = vi;
          }
        }
      }
      __syncthreads();
    }
  }

  // gather top-1024 (entries >= TOPK are inert padding)
  if (tid < KPAD) {
    const int i = tid;
    float score = -skey[i];
    unsigned v  = sval[i];
    int a   = (int)(v & 16383u);
    int lab = (int)(v >> 14);
    float bx0 = 0.f, by0 = 0.f, bx1 = 0.f, by1 = 0.f;
    float ox0, oy0, ox1, oy1;
    if (i < TOPK) {
      float x = P[0 * NA + a] * IMGSZ, y = P[1 * NA + a] * IMGSZ;
      float w = P[2 * NA + a] * IMGSZ, h = P[3 * NA + a] * IMGSZ;
      bx0 = x - 0.5f * w; by0 = y - 0.5f * h;
      bx1 = x + 0.5f * w; by1 = y + 0.5f * h;
      float off = (float)lab * MAXWH;
      ox0 = bx0 + off; oy0 = by0 + off; ox1 = bx1 + off; oy1 = by1 + off;
    } else {
      score = -1e30f; lab = 0;
      float far = 1.0e8f + (float)i * 1.0e5f;   // mutually disjoint, never suppresses
      ox0 = far; oy0 = far; ox1 = far + 1.f; oy1 = far + 1.f;
    }
    float* base = ws + (size_t)b * WS_PER_IMG;
    base[i] = score;
    ((int*)(base + KPAD))[i] = lab;
    float* tb = base + 2 * KPAD + i * 4;
    tb[0] = bx0; tb[1] = by0; tb[2] = bx1; tb[3] = by1;
    float* ob = base + 6 * KPAD + i * 4;
    ob[0] = ox0; ob[1] = oy0; ob[2] = ox1; ob[3] = oy1;
  }
}

// ------------------------------------------------------------------
// Kernel 2: per image — async-load boxes to LDS, WMMA-tiled pairwise
// suppression matrix (area_i+area_j via v_wmma_f32_16x16x4_f32),
// division-free IoU threshold (inter > t*union, union always > 0),
// wave-0 greedy keep over bit-words, thread-0 compaction to outputs.
// ------------------------------------------------------------------
__global__ __launch_bounds__(1024) void k_nms_out(const float* __restrict__ ws,
                                                  float* __restrict__ out) {
  __shared__ unsigned sup[KPAD * 32];  // 128 KB suppression bit-matrix
  __shared__ float4   sbox[KPAD];      //  16 KB class-offset boxes
  __shared__ float    sarea[KPAD];     //   4 KB areas
  __shared__ unsigned keepw[32];

  const int b    = blockIdx.x;
  const int tid  = threadIdx.x;
  const int lane = tid & 31;
  const int wave = tid >> 5;
  const float* __restrict__ base = ws + (size_t)b * WS_PER_IMG;

  for (int i = tid; i < KPAD * 32; i += 1024) sup[i] = 0u;

  // --- async global -> LDS staging of offset boxes (ASYNCcnt path) ---
  {
    const float* src = base + 6 * KPAD;             // SGPR base (uniform)
    unsigned voff = (unsigned)tid * 16u;            // per-lane byte offset
    unsigned ldsa = (unsigned)(unsigned long long)(&sbox[tid]); // LDS byte addr
    asm volatile("global_load_async_to_lds_b128 %0, %1, %2"
                 :: "v"(ldsa), "v"(voff), "s"(src) : "memory");
    asm volatile("s_wait_asynccnt 0" ::: "memory");
  }
  __syncthreads();

  {
    float4 bb = sbox[tid];
    sarea[tid] = (bb.z - bb.x) * (bb.w - bb.y);
  }
  __syncthreads();

  // --- WMMA-tiled suppression matrix: D[m][n] = area_i[m] + area_j[n] ---
  const int l15   = lane & 15;
  const bool loHalf = (lane < 16);
  for (int t = wave; t < NTILES; t += 32) {
    // decode flattened upper-triangular tile index -> (ti, tj)
    int tt = t, ti = 0;
    while (tt >= NT - ti) { tt -= NT - ti; ++ti; }
    int tj = ti + tt;
    int i0 = ti * 16, j0 = tj * 16;

    // A (16x4 f32): col0 = area_i, col1 = ones, K=2..3 zero (ISA layout:
    // lanes 0-15 carry M rows, VGPR0=K0, VGPR1=K1; lanes 16-31 = K2/K3 -> 0)
    v2f a, bv;
    a.x  = loHalf ? sarea[i0 + l15] : 0.f;
    a.y  = loHalf ? 1.f             : 0.f;
    // B (4x16 f32): row0 = ones, row1 = area_j (row-striped across lanes)
    bv.x = loHalf ? 1.f             : 0.f;
    bv.y = loHalf ? sarea[j0 + l15] : 0.f;
    v8f c = {0.f, 0.f, 0.f, 0.f, 0.f, 0.f, 0.f, 0.f};
    v8f d = __builtin_amdgcn_wmma_f32_16x16x4_f32(false, a, false, bv,
                                                  (short)0, c, false, false);

    const int jg = j0 + l15;                 // this lane's column
    const float4 bj = sbox[jg];
    const int mbase = (lane >> 4) * 8;       // D reg r -> row r (+8 hi half)
    #pragma unroll
    for (int r = 0; r < 8; ++r) {
      int ig = i0 + mbase + r;
      float4 bi = sbox[ig];
      float iw = fminf(bi.z, bj.z) - fmaxf(bi.x, bj.x);
      float ih = fminf(bi.w, bj.w) - fmaxf(bi.y, bj.y);
      float inter = fmaxf(iw, 0.f) * fmaxf(ih, 0.f);
      // iou > t  <=>  inter > t * (areaSum - inter + eps)   [union > 0]
      float rhs = IOU_T * (d[r] - inter + 1e-7f);
      if ((jg > ig) && (inter > rhs))
        atomicOr(&sup[ig * 32 + (jg >> 5)], 1u << (jg & 31));
    }
  }
  __syncthreads();

  // --- greedy keep: one wave, 32 bit-words across 32 lanes ---
  if (wave == 0) {
    volatile unsigned* kv = keepw;
    kv[lane] = 0xFFFFFFFFu;
    for (int i = 0; i < TOPK; ++i) {
      unsigned w = kv[i >> 5];                 // uniform read
      if ((w >> (i & 31)) & 1u)
        kv[lane] &= ~sup[i * 32 + lane];
    }
  }
  __syncthreads();

  // --- compaction + output (scores already descending => first 300 kept) ---
  if (tid == 0) {
    const int*   tl = (const int*)(base + KPAD);
    const float* tb = base + 2 * KPAD;
    float*     boxes_out  = out;
    float*     scores_out = out + (size_t)NB * MAXDET * 4;
    long long* labels_out = (long long*)(out + (size_t)NB * MAXDET * 5);
    int cnt = 0;
    for (int i = 0; i < TOPK && cnt < MAXDET; ++i) {
      bool kp = (keepw[i >> 5] >> (i & 31)) & 1u;
      float s = base[i];
      if (kp && s > CONF_T) {
        int o = b * MAXDET + cnt;
        boxes_out[o * 4 + 0] = tb[i * 4 + 0];
        boxes_out[o * 4 + 1] = tb[i * 4 + 1];
        boxes_out[o * 4 + 2] = tb[i * 4 + 2];
        boxes_out[o * 4 + 3] = tb[i * 4 + 3];
        scores_out[o] = s;
        labels_out[o] = (long long)d_coco[tl[i]];
        ++cnt;
      }
    }
    for (; cnt < MAXDET; ++cnt) {
      int o = b * MAXDET + cnt;
      boxes_out[o * 4 + 0] = 0.f; boxes_out[o * 4 + 1] = 0.f;
      boxes_out[o * 4 + 2] = 0.f; boxes_out[o * 4 + 3] = 0.f;
      scores_out[o] = 0.f;
      labels_out[o] = 0ll;
    }
  }
}

extern "C" void kernel_launch(void* const* d_in, const int* in_sizes, int n_in,
                              void* d_out, int out_size, void* d_ws, size_t ws_size,
                              hipStream_t stream) {
  const float* preds = (const float*)d_in[0];
  float* ws  = (float*)d_ws;     // needs 32*10240*4 B ~= 1.31 MB
  float* out = (float*)d_out;
  (void)in_sizes; (void)n_in; (void)out_size; (void)ws_size;

  k_score_sort<<<NB, 1024, 0, stream>>>(preds, ws);
  k_nms_out  <<<NB, 1024, 0, stream>>>(ws, out);
}